// SpikingRaphe_38869454029218
// MI455X (gfx1250) — compile-verified
//
#include <hip/hip_runtime.h>

// ---------------------------------------------------------------------------
// SpikingRaphe on MI455X (gfx1250, wave32).
// Memory-bound streaming kernel: ~736 MB of one-shot fp32 traffic -> ~32us
// floor at 23.3 TB/s. Strategy:
//   * 4 neurons / thread, 128-bit non-temporal global loads (no L2 pollution,
//     data is read exactly once and exceeds the 192MB L2 anyway)
//   * fully unrolled T=20 recurrence, fmaf/select-only inner math
//   * wave32 reduction that routes through V_WMMA_F32_16X16X4_F32
//     (A = ones -> D rows hold column sums s_n + s_{n+16}; correct 32->16
//     lane reduction) + 4 shfl_xor steps + LDS across the 8 waves
//   * one fp32 partial per block into d_ws; a single-block finalize kernel
//     reduces partials in double (deterministic -- no fp atomics) and emits
//     the 5 output scalars.
// ---------------------------------------------------------------------------

typedef __attribute__((ext_vector_type(4))) float vf4;
typedef __attribute__((ext_vector_type(2))) float v2f;
typedef __attribute__((ext_vector_type(8))) float v8f;

__device__ __forceinline__ void izh_step(float& v, float& u, float& r,
                                         float n, float i_const) {
    // I = i_base + i_tonic + 0.5*n  (i_const folds i_base + i_tonic)
    float I = fmaf(0.5f, n, i_const);
    // v += 0.04 v^2 + 5 v + 140 - u + I
    float poly = fmaf(0.04f, v * v, fmaf(5.0f, v, 140.0f));
    v = v + (poly - u + I);
    // u += A * (B*v_new - u)
    u = fmaf(0.02f, fmaf(0.2f, v, -u), u);
    bool sp = (v >= 30.0f);
    v = sp ? -65.0f : v;
    u = sp ? (u + 8.0f) : u;
    // r = 0.9 r + 0.1 * spike
    r = fmaf(0.9f, r, sp ? 0.1f : 0.0f);
}

__global__ void __launch_bounds__(256)
izh_pop_kernel(const vf4* __restrict__ noise,   // (T, nvec4) rows of float4
               const vf4* __restrict__ v_in,
               const vf4* __restrict__ u_in,
               const vf4* __restrict__ r_in,
               float* __restrict__ partials,    // one per block
               int nvec4, float i_const) {
    constexpr int T_STEPS = 20;
    const int i4 = blockIdx.x * 256 + threadIdx.x;

    float s = 0.0f;
    if (i4 < nvec4) {
        float v[4], u[4], r[4];
        {
            vf4 tv = __builtin_nontemporal_load(v_in + i4);
            vf4 tu = __builtin_nontemporal_load(u_in + i4);
            vf4 tr = __builtin_nontemporal_load(r_in + i4);
            #pragma unroll
            for (int j = 0; j < 4; ++j) { v[j] = tv[j]; u[j] = tu[j]; r[j] = tr[j]; }
        }
        const vf4* np = noise + i4;
        #pragma unroll
        for (int t = 0; t < T_STEPS; ++t) {
            vf4 n = __builtin_nontemporal_load(np);   // streaming, read-once
            np += nvec4;                               // next timestep row
            #pragma unroll
            for (int j = 0; j < 4; ++j)
                izh_step(v[j], u[j], r[j], n[j], i_const);
        }
        s = (r[0] + r[1]) + (r[2] + r[3]);
    }

    // ---- wave32 reduction, first stage through the matrix pipe ----
    // A = ones(16x4); B vgpr0 = s (K=0 row from lanes 0-15, K=2 row from
    // lanes 16-31), B vgpr1 = 0.  D[m][n] = s_n + s_{n+16} for every row m,
    // so D-vgpr0 holds the 32->16 partial sums in all lanes.
    // All 256 threads reach this point (no early return) => EXEC is all ones,
    // as required for WMMA.
    v2f a; a[0] = 1.0f; a[1] = 1.0f;
    v2f b; b[0] = s;    b[1] = 0.0f;
    v8f c = {};
    c = __builtin_amdgcn_wmma_f32_16x16x4_f32(
            /*neg_a=*/false, a, /*neg_b=*/false, b,
            /*c_mod=*/(short)0, c, /*reuse_a=*/false, /*reuse_b=*/false);
    float t = c[0];
    // finish within each 16-lane half (both halves hold the same t_n vector)
    t += __shfl_xor(t, 1, 32);
    t += __shfl_xor(t, 2, 32);
    t += __shfl_xor(t, 4, 32);
    t += __shfl_xor(t, 8, 32);

    __shared__ float wsum[8];
    const int wave = threadIdx.x >> 5;
    if ((threadIdx.x & 31) == 0) wsum[wave] = t;
    __syncthreads();
    if (threadIdx.x == 0) {
        float bs = 0.0f;
        #pragma unroll
        for (int w = 0; w < 8; ++w) bs += wsum[w];
        partials[blockIdx.x] = bs;
    }
}

__global__ void __launch_bounds__(256)
finalize_kernel(const float* __restrict__ p_dr, int nb_dr,
                const float* __restrict__ p_mr, int nb_mr,
                float* __restrict__ out, int n_dr, int n_mr) {
    double sdr = 0.0, smr = 0.0;
    for (int i = threadIdx.x; i < nb_dr; i += 256) sdr += (double)p_dr[i];
    for (int i = threadIdx.x; i < nb_mr; i += 256) smr += (double)p_mr[i];
    #pragma unroll
    for (int m = 16; m >= 1; m >>= 1) {
        sdr += __shfl_xor(sdr, m, 32);
        smr += __shfl_xor(smr, m, 32);
    }
    __shared__ double sh_dr[8], sh_mr[8];
    const int wave = threadIdx.x >> 5;
    if ((threadIdx.x & 31) == 0) { sh_dr[wave] = sdr; sh_mr[wave] = smr; }
    __syncthreads();
    if (threadIdx.x == 0) {
        double d = 0.0, m2 = 0.0;
        #pragma unroll
        for (int w = 0; w < 8; ++w) { d += sh_dr[w]; m2 += sh_mr[w]; }
        float dr_mean = (float)(d  / (double)n_dr);
        float mr_mean = (float)(m2 / (double)n_mr);
        float raw = fminf(fmaxf(dr_mean * 8.0f, 0.0f), 1.0f);
        float ema = 0.9f * 0.5f + 0.1f * raw;           // _ht5_ema starts 0.5
        float ht5 = fminf(fmaxf(ema, 0.05f), 0.95f);
        float gain = 1.0f - 0.3f * ht5;
        out[0] = ht5;
        out[1] = dr_mean;
        out[2] = mr_mean;
        out[3] = gain;
        out[4] = ht5;
    }
}

extern "C" void kernel_launch(void* const* d_in, const int* in_sizes, int n_in,
                              void* d_out, int out_size, void* d_ws, size_t ws_size,
                              hipStream_t stream) {
    const float* noise_dr = (const float*)d_in[0];   // (T, N_DR)
    const float* noise_mr = (const float*)d_in[1];   // (T, N_MR)
    const float* v_dr     = (const float*)d_in[2];
    const float* u_dr     = (const float*)d_in[3];
    const float* r_dr     = (const float*)d_in[4];
    const float* v_mr     = (const float*)d_in[5];
    const float* u_mr     = (const float*)d_in[6];
    const float* r_mr     = (const float*)d_in[7];

    const int n_dr = in_sizes[2];
    const int n_mr = in_sizes[5];
    const int nv4_dr = n_dr >> 2;                    // 6,000,000 / 4
    const int nv4_mr = n_mr >> 2;                    // 2,000,000 / 4
    const int nb_dr = (nv4_dr + 255) / 256;
    const int nb_mr = (nv4_mr + 255) / 256;

    float* p_dr = (float*)d_ws;                      // nb_dr partials
    float* p_mr = p_dr + nb_dr;                      // nb_mr partials

    // Host-folded drives: base = 3 + 0.7*5 = 6.5
    //   DR: max(0, 6.5)      + (-0.5) = 6.00
    //   MR: max(0, 6.5*0.7)  + (-1.0) = 3.55
    izh_pop_kernel<<<nb_dr, 256, 0, stream>>>(
        (const vf4*)noise_dr, (const vf4*)v_dr, (const vf4*)u_dr,
        (const vf4*)r_dr, p_dr, nv4_dr, 6.00f);
    izh_pop_kernel<<<nb_mr, 256, 0, stream>>>(
        (const vf4*)noise_mr, (const vf4*)v_mr, (const vf4*)u_mr,
        (const vf4*)r_mr, p_mr, nv4_mr, 3.55f);

    finalize_kernel<<<1, 256, 0, stream>>>(p_dr, nb_dr, p_mr, nb_mr,
                                           (float*)d_out, n_dr, n_mr);
}